// Apply2DTform_89292370084476
// MI455X (gfx1250) — compile-verified
//
#include <hip/hip_runtime.h>

typedef __attribute__((ext_vector_type(2))) float v2f;
typedef __attribute__((ext_vector_type(8))) float v8f;

#define IMG_H 512
#define IMG_W 512
#define IMG_C 3
#define NBATCH 32

// One wave (32 lanes) per 16x16 output tile.
// Coordinates for the whole tile are produced by two V_WMMA_F32_16X16X4_F32
// ops (rank-2 affine form); the rest is the memory-bound gather/blend.
__global__ __launch_bounds__(256) void stn_resample_kernel(
    const float* __restrict__ img,   // (32,512,512,3) f32
    const float* __restrict__ tf,    // (32,6) f32
    float* __restrict__ out)         // (32,512,512,3) f32
{
    const int lane = threadIdx.x & 31;
    const int wave = threadIdx.x >> 5;
    const int tile = blockIdx.x * 8 + wave;   // 0..32767
    const int b    = tile >> 10;              // / 1024 tiles per image
    const int t    = tile & 1023;
    const int i0   = (t >> 5) << 4;           // tile row base
    const int j0   = (t & 31) << 4;           // tile col base

    // Affine params (wave-uniform)
    const float t0 = tf[b*6+0], t1 = tf[b*6+1], t2 = tf[b*6+2];
    const float t3 = tf[b*6+3], t4 = tf[b*6+4], t5 = tf[b*6+5];
    // x = t0*i + t1*j + Cx ; y = t2*i + t3*j + Cy   (255.5*2/511 == 1 exactly)
    const float Cx = 255.5f * (1.0f + t4 - t0 - t1);
    const float Cy = 255.5f * (1.0f + t5 - t2 - t3);

    const bool lo = lane < 16;
    const int  n  = lane & 15;

    // A (16x4 f32): lanes 0-15 hold K=0 (v0) / K=1 (v1) for row M=lane;
    // lanes 16-31 hold K=2/K=3 which are zero coefficients here.
    v2f A;
    A.x = lo ? 1.0f : 0.0f;                       // K=0: constant term selector
    A.y = lo ? (float)(i0 + n) : 0.0f;            // K=1: row index
    // B (4x16 f32): row K=0 = t1*j + C ; row K=1 = t0 (row-slope); K=2,3 = 0
    v2f Bx, By;
    Bx.x = lo ? (t1 * (float)(j0 + n) + Cx) : 0.0f;
    Bx.y = lo ? t0 : 0.0f;
    By.x = lo ? (t3 * (float)(j0 + n) + Cy) : 0.0f;
    By.y = lo ? t2 : 0.0f;

    v8f zero = {};
    // D[i][j] = (t1*j + C) + i*t0  -> full 16x16 coordinate tile in 8 VGPRs
    v8f xd = __builtin_amdgcn_wmma_f32_16x16x4_f32(false, A, false, Bx,
                                                   (short)0, zero, false, false);
    v8f yd = __builtin_amdgcn_wmma_f32_16x16x4_f32(false, A, false, By,
                                                   (short)0, zero, false, false);

    const int j     = j0 + n;                     // this lane's output column
    const int iHalf = lo ? 0 : 8;                 // D layout: lanes>=16 -> rows r+8
    const float* __restrict__ imgB = img + (size_t)b * (IMG_H * IMG_W * IMG_C);
    float*       __restrict__ outB = out + (size_t)b * (IMG_H * IMG_W * IMG_C);

#pragma unroll
    for (int r = 0; r < 8; ++r) {
        const int   i  = i0 + r + iHalf;
        const float xf = xd[r];
        const float yf = yd[r];

        // Reference quirk: ROUND (nearest-even, == jnp.round) for base corner
        int x0 = (int)rintf(xf);
        int y0 = (int)rintf(yf);
        int x1 = x0 + 1;
        int y1 = y0 + 1;
        // clip to [0, 512]  (512 == zero-pad row/col of Imgp)
        x0 = min(max(x0, 0), 512); x1 = min(max(x1, 0), 512);
        y0 = min(max(y0, 0), 512); y1 = min(max(y1, 0), 512);

        const float fx0 = (float)x0, fx1 = (float)x1;
        const float fy0 = (float)y0, fy1 = (float)y1;
        const float w00 = (fx1 - xf) * (fy1 - yf);
        const float w01 = (fx1 - xf) * (yf - fy0);
        const float w10 = (xf - fx0) * (fy1 - yf);
        const float w11 = (xf - fx0) * (yf - fy0);

        // Branchless padded fetch: clamp address into the real image, mask
        // the value to zero when the (clipped) index hits the pad (==512).
        const int xs0 = min(x0, IMG_H - 1), xs1 = min(x1, IMG_H - 1);
        const int ys0 = min(y0, IMG_W - 1), ys1 = min(y1, IMG_W - 1);
        const float m00 = (x0 < IMG_H && y0 < IMG_W) ? 1.0f : 0.0f;
        const float m01 = (x0 < IMG_H && y1 < IMG_W) ? 1.0f : 0.0f;
        const float m10 = (x1 < IMG_H && y0 < IMG_W) ? 1.0f : 0.0f;
        const float m11 = (x1 < IMG_H && y1 < IMG_W) ? 1.0f : 0.0f;

        const float* p00 = imgB + ((size_t)xs0 * IMG_W + ys0) * IMG_C;
        const float* p01 = imgB + ((size_t)xs0 * IMG_W + ys1) * IMG_C;
        const float* p10 = imgB + ((size_t)xs1 * IMG_W + ys0) * IMG_C;
        const float* p11 = imgB + ((size_t)xs1 * IMG_W + ys1) * IMG_C;

        float* po = outB + ((size_t)i * IMG_W + j) * IMG_C;
#pragma unroll
        for (int c = 0; c < IMG_C; ++c) {
            po[c] = w00 * (m00 * p00[c]) + w01 * (m01 * p01[c])
                  + w10 * (m10 * p10[c]) + w11 * (m11 * p11[c]);
        }
    }
}

extern "C" void kernel_launch(void* const* d_in, const int* in_sizes, int n_in,
                              void* d_out, int out_size, void* d_ws, size_t ws_size,
                              hipStream_t stream) {
    (void)in_sizes; (void)n_in; (void)out_size; (void)d_ws; (void)ws_size;
    const float* img = (const float*)d_in[0];   // (32,512,512,3) f32
    const float* tf  = (const float*)d_in[1];   // (32,6) f32
    float* out = (float*)d_out;

    // 32768 tiles (16x16) / 8 waves per 256-thread block = 4096 blocks.
    dim3 grid(4096), block(256);
    hipLaunchKernelGGL(stn_resample_kernel, grid, block, 0, stream, img, tf, out);
}